// Hadmard_24240795419355
// MI455X (gfx1250) — compile-verified
//
#include <hip/hip_runtime.h>

typedef float f4 __attribute__((ext_vector_type(4)));

constexpr int THREADS = 256;           // 8 waves (wave32)
constexpr int EDGES_PER_BLOCK = 128;   // 1 KB of 32-bit indices staged in LDS
constexpr int D = 64;                  // floats per node row (256 B)

__global__ __launch_bounds__(THREADS)
void u_mul_v_kernel(const float* __restrict__ h,
                    const long long* __restrict__ src,
                    const long long* __restrict__ dst,
                    float* __restrict__ out,
                    int n_edges)
{
    // [0..127] = src indices (low dword), [128..255] = dst indices
    __shared__ int s_idx[2 * EDGES_PER_BLOCK];

    const int tid = threadIdx.x;
    const long long ebase = (long long)blockIdx.x * EDGES_PER_BLOCK;

    // ---- Stage low 32 bits of this block's edge indices into LDS (async) ----
    long long e_stage = ebase + (tid & (EDGES_PER_BLOCK - 1));
    if (e_stage >= n_edges) e_stage = (long long)n_edges - 1;  // clamp; value unused
    const long long* gq = (tid < EDGES_PER_BLOCK) ? (src + e_stage)
                                                  : (dst + e_stage);
    const int* gsrc = (const int*)gq;   // little-endian low dword of int64 index

#if __has_builtin(__builtin_amdgcn_global_load_async_to_lds_b32)
    __builtin_amdgcn_global_load_async_to_lds_b32(
        (int*)gsrc, &s_idx[tid], /*offset=*/0, /*cpol=*/0);
  #if __has_builtin(__builtin_amdgcn_s_wait_asynccnt)
    __builtin_amdgcn_s_wait_asynccnt(0);
  #else
    asm volatile("s_wait_asynccnt 0" ::: "memory");
  #endif
#else
    s_idx[tid] = *gsrc;   // fallback: plain LDS store
#endif
    __syncthreads();

    // ---- Gather-multiply-store: 16 lanes cover one 256B row with float4 ----
    const int sub   = tid >> 4;          // edge slot within a group of 16 edges
    const int cbase = (tid & 15) * 4;    // float element offset within the row

    float* __restrict__ outb = out + ebase * D;   // uniform 64-bit base, once

    if (ebase + EDGES_PER_BLOCK <= n_edges) {
        // Fast path: full block, no per-iteration guards, pure 32-bit offsets
#pragma unroll
        for (int it = 0; it < EDGES_PER_BLOCK / 16; ++it) {
            const int el = it * 16 + sub;                     // local edge 0..127
            const int si = s_idx[el];                         // LDS broadcast
            const int di = s_idx[EDGES_PER_BLOCK + el];

            const f4 a = *(const f4*)(h + si * D + cbase);    // temporal: h lives in L2
            const f4 b = *(const f4*)(h + di * D + cbase);
            const f4 r = a * b;

            // Streaming output: non-temporal b128, constant 4KB stride per iter
            __builtin_nontemporal_store(r, (f4*)(outb + el * D + cbase));
        }
    } else {
        // Tail path: per-edge guard (only last block, only if E % 128 != 0)
        for (int it = 0; it < EDGES_PER_BLOCK / 16; ++it) {
            const int el = it * 16 + sub;
            if (ebase + el >= n_edges) break;
            const int si = s_idx[el];
            const int di = s_idx[EDGES_PER_BLOCK + el];

            const f4 a = *(const f4*)(h + si * D + cbase);
            const f4 b = *(const f4*)(h + di * D + cbase);
            const f4 r = a * b;
            __builtin_nontemporal_store(r, (f4*)(outb + el * D + cbase));
        }
    }
}

extern "C" void kernel_launch(void* const* d_in, const int* in_sizes, int n_in,
                              void* d_out, int out_size, void* d_ws, size_t ws_size,
                              hipStream_t stream) {
    const float*     h   = (const float*)d_in[0];
    const long long* src = (const long long*)d_in[1];
    const long long* dst = (const long long*)d_in[2];
    float*           out = (float*)d_out;

    const int n_edges = in_sizes[1];   // 800000
    const int blocks  = (n_edges + EDGES_PER_BLOCK - 1) / EDGES_PER_BLOCK;  // 6250

    u_mul_v_kernel<<<blocks, THREADS, 0, stream>>>(h, src, dst, out, n_edges);
}